// KernelConv_86414741995617
// MI455X (gfx1250) — compile-verified
//
#include <hip/hip_runtime.h>

typedef __attribute__((ext_vector_type(2))) float v2f;
typedef __attribute__((ext_vector_type(8))) float v8f;

#define NB 160
#define NPATH 1536
#define HDIM 64

// padded LDS strides (avoid bank conflicts, keep float2 8B alignment for hS)
#define HS_STRIDE 66
#define F0_STRIDE 17
#define F1_STRIDE 49
#define SC_STRIDE 4

__global__ __launch_bounds__(256) void e3nn_tp_conv_kernel(
    const float* __restrict__ features, const float* __restrict__ geometry,
    const float* __restrict__ mask, const float* __restrict__ W1,
    const float* __restrict__ b1, const float* __restrict__ W2,
    const float* __restrict__ b2, float* __restrict__ out)
{
  extern __shared__ float smem[];
  float* hS   = smem;                          // [160][66]  hidden layer (GEMM A)
  float* a0S  = hS  + NB * HS_STRIDE;          // [160][17]  t00 * f0        (grp 0)
  float* a1S  = a0S + NB * F0_STRIDE;          // [160][17]  nc01/s3*(Y1.f1) (grp 1)
  float* c0S  = a1S + NB * F0_STRIDE;          // [160][17]  t10 * f0        (grp 2)
  float* g0S  = c0S + NB * F0_STRIDE;          // [160][49]  s0 * f1         (grp 3,k=0)
  float* g1S  = g0S + NB * F1_STRIDE;          // [160][49]  eps coupling    (grp 3,k=1)
  float* g2S  = g1S + NB * F1_STRIDE;          // [160][49]  l_f=2 coupling  (grp 3,k=2)
  float* scS  = g2S + NB * F1_STRIDE;          // [160][4]   {Y1y, Y1z, Y1x}
  float* outS = scS + NB * SC_STRIDE;          // [64]       output accumulators

  const int tid = threadIdx.x;
  const int z  = blockIdx.x / 160;
  const int ai = blockIdx.x % 160;

  if (tid < 64) outS[tid] = 0.f;

  // ---------------- Phase 1: per-b geometry, SH, radial hidden layer ----------------
  if (tid < NB) {
    const int b = tid;
    const float* g = geometry + (((size_t)z * 160 + ai) * 160 + b) * 3;
    float gx = g[0], gy = g[1], gz = g[2];
    float r = sqrtf(gx*gx + gy*gy + gz*gz);
    float inv = (r > 0.f) ? 1.f / r : 0.f;
    float x = gx*inv, y = gy*inv, zz = gz*inv;

    const float c1 = 0.4886025119029199f;
    float Y1y = c1*y, Y1z = c1*zz, Y1x = c1*x;
    const float k1 = 1.0925484305920792f, k2 = 0.31539156525252005f, k3 = 0.5462742152960396f;
    float Y2_0 = k1*x*y, Y2_1 = k1*y*zz, Y2_2 = k2*(3.f*zz*zz - 1.f),
          Y2_3 = k1*x*zz, Y2_4 = k3*(x*x - y*y);

    bool iz = (r == 0.f);
    const float s4pi = 3.5449077018110318f;      // sqrt(4*pi)
    const float sr3  = 1.7320508075688772f;
    float nc00 = iz ? s4pi * 0.25f       : s4pi * 0.17677669529663687f;  // /sqrt(32)
    float nc01 = iz ? sr3 * s4pi * 0.25f : sr3 * s4pi * 0.17677669529663687f;
    float nc10 = iz ? s4pi * 0.25f       : s4pi * 0.125f;                // /8
    float nc11 = iz ? sr3 * s4pi * 0.25f : sr3 * s4pi * 0.125f;

    const float Y0     = 0.28209479177387814f;
    const float inv_s3 = 0.5773502691896258f;
    const float inv_s6 = 0.4082482904638630f;
    float t00  = nc00 * Y0;          // group A scale
    float a1sc = nc01 * inv_s3;      // group B scale
    float t10  = nc10 * inv_s3;      // group C scale
    float s0   = nc11 * Y0 * inv_s3; // group D, l_f=0 scale
    float g1sc = nc11 * inv_s6;      // group D, l_f=1 scale
    float g2sc = nc11;               // group D, l_f=2 scale (w3j normalized)

    scS[b*SC_STRIDE+0] = Y1y;  scS[b*SC_STRIDE+1] = Y1z;  scS[b*SC_STRIDE+2] = Y1x;

    // symmetric 3x3 of w3j(1,1,2) contracted with Y2 (index order y,z,x)
    const float q   = 0.31622776601683794f;  // 1/sqrt(10)
    const float q30 = 0.18257418583505536f;  // 1/sqrt(30)
    float M00 = -q30*Y2_2 - q*Y2_4;
    float M11 =  2.f*q30*Y2_2;
    float M22 = -q30*Y2_2 + q*Y2_4;
    float M01 =  q*Y2_1, M02 = q*Y2_0, M12 = q*Y2_3;

    const float* f = features + ((size_t)z * 160 + b) * 64;
    #pragma unroll 4
    for (int v = 0; v < 16; ++v) {
      float f0v = f[v];
      float fy = f[16 + v*3 + 0], fz = f[16 + v*3 + 1], fx = f[16 + v*3 + 2];
      a0S[b*F0_STRIDE + v] = t00 * f0v;
      c0S[b*F0_STRIDE + v] = t10 * f0v;
      a1S[b*F0_STRIDE + v] = a1sc * (Y1y*fy + Y1z*fz + Y1x*fx);
      g0S[b*F1_STRIDE + v*3 + 0] = s0 * fy;
      g0S[b*F1_STRIDE + v*3 + 1] = s0 * fz;
      g0S[b*F1_STRIDE + v*3 + 2] = s0 * fx;
      // eps_{ijk} f_j Y_k  (cross product in (y,z,x) index space)
      g1S[b*F1_STRIDE + v*3 + 0] = g1sc * (fz*Y1x - fx*Y1z);
      g1S[b*F1_STRIDE + v*3 + 1] = g1sc * (fx*Y1y - fy*Y1x);
      g1S[b*F1_STRIDE + v*3 + 2] = g1sc * (fy*Y1z - fz*Y1y);
      g2S[b*F1_STRIDE + v*3 + 0] = g2sc * (M00*fy + M01*fz + M02*fx);
      g2S[b*F1_STRIDE + v*3 + 1] = g2sc * (M01*fy + M11*fz + M12*fx);
      g2S[b*F1_STRIDE + v*3 + 2] = g2sc * (M02*fy + M12*fz + M22*fx);
    }
    // radial MLP hidden layer: h = relu(r*W1 + b1)
    #pragma unroll 4
    for (int t = 0; t < HDIM; ++t)
      hS[b*HS_STRIDE + t] = fmaxf(r * W1[t] + b1[t], 0.f);
  }
  __syncthreads();

  // ---------------- Phase 2: fused WMMA GEMM (R = h @ W2 + b2) + consumption --------
  const int lane = tid & 31;
  const int wave = tid >> 5;
  const int lo = lane & 15;   // N (path column) / M (b row) within tile
  const int hi = lane >> 4;   // half-wave select

  for (int nt = wave; nt < 96; nt += 8) {
    const int p0 = nt * 16;
    const int pcol = p0 + lo;

    if (nt + 8 < 96)  // hint next W2 tile toward the caches
      __builtin_prefetch(&W2[(size_t)(2*hi) * NPATH + p0 + 128 + lo], 0, 1);

    // B fragments for all 16 K-steps, held across M tiles.
    // Layout (ISA 7.12.2, f32 4x16 B): VGPR0 = rows K, K+2 ; VGPR1 = rows K+1, K+3
    v2f Bf[16];
    #pragma unroll
    for (int ks = 0; ks < 16; ++ks) {
      int row = ks * 4 + 2 * hi;
      Bf[ks].x = W2[(size_t)row       * NPATH + pcol];
      Bf[ks].y = W2[(size_t)(row + 1) * NPATH + pcol];
    }
    float b2p = b2[pcol];

    // Branchless per-lane table selection (wave-uniform grp; per-lane kf -> address select)
    int grp, u;
    const float* tab1 = a0S;   // grp 0/1/2: one weight per (b) at stride F0_STRIDE
    const float* tab3 = g0S;   // grp 3: three weights per (b) at stride F1_STRIDE
    if (p0 < 768) {
      grp = p0 >> 8; u = (p0 & 255) >> 4;
      tab1 = (grp == 0 ? a0S : (grp == 1 ? a1S : c0S)) + lo;
    } else {
      grp = 3;
      int idx = pcol - 768;
      u = idx / 48;
      int rem = idx % 48, v = rem / 3, kf = rem % 3;
      tab3 = (kf == 0 ? g0S : (kf == 1 ? g1S : g2S)) + v * 3;
    }

    float acc0 = 0.f, acc1 = 0.f, acc2 = 0.f;

    for (int mt = 0; mt < 10; ++mt) {
      const int b0 = mt * 16;
      v8f c = {0.f, 0.f, 0.f, 0.f, 0.f, 0.f, 0.f, 0.f};
      const float* arow = &hS[(b0 + lo) * HS_STRIDE + 2 * hi];
      #pragma unroll
      for (int ks = 0; ks < 16; ++ks) {
        v2f av;                      // A layout: VGPR0 = K, VGPR1 = K+1 (lane-half selects K base)
        av.x = arow[ks * 4];
        av.y = arow[ks * 4 + 1];
        c = __builtin_amdgcn_wmma_f32_16x16x4_f32(false, av, false, Bf[ks],
                                                  (short)0, c, false, false);
      }
      // consume R tile: lane owns column p, regs r hold b rows b0 + r + 8*hi
      const int bbase = b0 + hi * 8;
      if (grp <= 1) {
        #pragma unroll
        for (int r = 0; r < 8; ++r)
          acc0 += (c[r] + b2p) * tab1[(bbase + r) * F0_STRIDE];
      } else if (grp == 2) {
        #pragma unroll
        for (int r = 0; r < 8; ++r) {
          int bb = bbase + r;
          float t = (c[r] + b2p) * tab1[bb * F0_STRIDE];
          acc0 += t * scS[bb*SC_STRIDE + 0];
          acc1 += t * scS[bb*SC_STRIDE + 1];
          acc2 += t * scS[bb*SC_STRIDE + 2];
        }
      } else {
        #pragma unroll
        for (int r = 0; r < 8; ++r) {
          float val = c[r] + b2p;
          const float* p = tab3 + (bbase + r) * F1_STRIDE;
          acc0 += val * p[0];
          acc1 += val * p[1];
          acc2 += val * p[2];
        }
      }
    }

    // cross-lane reduction (wave32) and accumulate into shared outputs
    #pragma unroll
    for (int m = 16; m >= 1; m >>= 1) {
      acc0 += __shfl_xor(acc0, m, 32);
      acc1 += __shfl_xor(acc1, m, 32);
      acc2 += __shfl_xor(acc2, m, 32);
    }
    if (lane == 0) {
      if (grp <= 1) {
        atomicAdd(&outS[u], acc0);
      } else {
        atomicAdd(&outS[16 + u*3 + 0], acc0);
        atomicAdd(&outS[16 + u*3 + 1], acc1);
        atomicAdd(&outS[16 + u*3 + 2], acc2);
      }
    }
  }
  __syncthreads();

  // ---------------- Phase 3: masked write ----------------
  if (tid < 64)
    out[(((size_t)z * 160) + ai) * 64 + tid] = outS[tid] * mask[z * 160 + ai];
}

extern "C" void kernel_launch(void* const* d_in, const int* in_sizes, int n_in,
                              void* d_out, int out_size, void* d_ws, size_t ws_size,
                              hipStream_t stream) {
  const float* features = (const float*)d_in[0];
  const float* geometry = (const float*)d_in[1];
  const float* mask     = (const float*)d_in[2];
  const float* W1       = (const float*)d_in[3];
  const float* b1       = (const float*)d_in[4];
  const float* W2       = (const float*)d_in[5];
  const float* b2       = (const float*)d_in[6];
  float* out = (float*)d_out;

  size_t lds = (size_t)(NB*HS_STRIDE + 3*NB*F0_STRIDE + 3*NB*F1_STRIDE +
                        NB*SC_STRIDE + 64) * sizeof(float);   // ~172 KB of 320 KB/WGP
  (void)in_sizes; (void)n_in; (void)out_size; (void)d_ws; (void)ws_size;
  e3nn_tp_conv_kernel<<<320, 256, lds, stream>>>(features, geometry, mask,
                                                 W1, b1, W2, b2, out);
}